// FeaturePropogation_60610578481729
// MI455X (gfx1250) — compile-verified
//
#include <hip/hip_runtime.h>
#include <hip/hip_bf16.h>
#include <math.h>

// Problem constants (from setup_inputs)
#define BATCH 2
#define NP1   16384   // N1 query points
#define NP2   4096    // N2 source points
#define C1CH  128
#define C2CH  256
#define C0    384     // C1+C2 (GEMM1 K)
#define OCH   256     // output channels of both blocks
#define BN_EPS 1e-5f

typedef float v2f __attribute__((ext_vector_type(2)));
typedef float v8f __attribute__((ext_vector_type(8)));

// ---------------------------------------------------------------------------
// Kernel 1: 3-NN search + inverse-distance weights.
// One lane per query point; all of p2 cached in LDS (48 KB, WGP has 320 KB).
// ---------------------------------------------------------------------------
__global__ void fp_knn_kernel(const float* __restrict__ p1,
                              const float* __restrict__ p2,
                              int* __restrict__ idx,
                              float* __restrict__ w) {
    __shared__ float sx[NP2];
    __shared__ float sy[NP2];
    __shared__ float sz[NP2];
    const int b = blockIdx.y;
    const float* p2b = p2 + (size_t)b * NP2 * 3;
    for (int j = threadIdx.x; j < NP2; j += 256) {
        sx[j] = p2b[j * 3 + 0];
        sy[j] = p2b[j * 3 + 1];
        sz[j] = p2b[j * 3 + 2];
    }
    __syncthreads();

    const int n = blockIdx.x * 256 + threadIdx.x;
    const float* q = p1 + ((size_t)b * NP1 + n) * 3;
    const float qx = q[0], qy = q[1], qz = q[2];

    float d0 = 3.4e38f, d1 = 3.4e38f, d2 = 3.4e38f;
    int   i0 = 0,       i1 = 0,       i2 = 0;
    for (int j = 0; j < NP2; ++j) {
        const float dx = qx - sx[j];
        const float dy = qy - sy[j];
        const float dz = qz - sz[j];
        const float d = fmaf(dx, dx, fmaf(dy, dy, dz * dz));
        if (d < d2) {
            if (d < d1) {
                d2 = d1; i2 = i1;
                if (d < d0) { d1 = d0; i1 = i0; d0 = d; i0 = j; }
                else        { d1 = d;  i1 = j; }
            } else { d2 = d; i2 = j; }
        }
    }
    float w0 = 1.0f / (d0 + 1e-16f);
    float w1 = 1.0f / (d1 + 1e-16f);
    float w2 = 1.0f / (d2 + 1e-16f);
    const float inv = 1.0f / (w0 + w1 + w2);
    const size_t base = ((size_t)b * NP1 + n) * 3;
    idx[base + 0] = i0; idx[base + 1] = i1; idx[base + 2] = i2;
    w[base + 0] = w0 * inv; w[base + 1] = w1 * inv; w[base + 2] = w2 * inv;
}

// ---------------------------------------------------------------------------
// Kernel 2: build X0 = concat([x1, knn_interpolate(x2)]) : (B, 384, N1)
// ---------------------------------------------------------------------------
__global__ void fp_interp_concat_kernel(const float* __restrict__ x1,
                                        const float* __restrict__ x2,
                                        const int* __restrict__ idx,
                                        const float* __restrict__ w,
                                        float* __restrict__ X0) {
    const int t = blockIdx.x * 256 + threadIdx.x;
    const int n = t & (NP1 - 1);
    const int c = (t / NP1) % C0;
    const int b = t / (NP1 * C0);
    float v;
    if (c < C1CH) {
        v = x1[((size_t)b * C1CH + c) * NP1 + n];
    } else {
        const int c2 = c - C1CH;
        const size_t g = ((size_t)b * NP1 + n) * 3;
        const float* row = x2 + ((size_t)b * C2CH + c2) * NP2;
        v = w[g + 0] * row[idx[g + 0]]
          + w[g + 1] * row[idx[g + 1]]
          + w[g + 2] * row[idx[g + 2]];
    }
    X0[((size_t)b * C0 + c) * NP1 + n] = v;
}

// ---------------------------------------------------------------------------
// Kernel 3: fp32 WMMA GEMM  Y[b] = W(OxC) * X[b](CxN1) + bias.
// FUSED: input transform x <- relu(s[k]*x + t[k]) (BN+ReLU of prev layer),
// resolved at compile time so the K-loop is branch-free.
// One wave computes a 64(O) x 32(N) strip: 4 O-tiles x 2 N-tiles (8 WMMA per
// K-step). All 8 fragment loads of a K-step are issued into DISTINCT
// registers before the WMMA burst so the scheduler can clause them and wait
// once, instead of a load->wait->wmma serial chain per fragment.
// 8 waves per block; EXEC is always all-ones (WMMA requirement).
// ---------------------------------------------------------------------------
template <bool FUSED>
__global__ void fp_gemm_wmma_kernel(const float* __restrict__ W,
                                    const float* __restrict__ bias,
                                    const float* __restrict__ X,
                                    float* __restrict__ Y,
                                    const float* __restrict__ s,
                                    const float* __restrict__ t,
                                    int C) {
    const int lane   = threadIdx.x & 31;
    const int wave   = threadIdx.x >> 5;
    const int half   = lane >> 4;      // 0: K={0,1}, 1: K={2,3}
    const int lanelo = lane & 15;      // M / N index within tile
    const int n0 = (blockIdx.x * 8 + wave) * 32;
    const int ob = blockIdx.y * 64;
    const int b  = blockIdx.z;

    // Per-lane base pointers (hoist all address math out of the K-loop).
    const float* Xc0 = X + (size_t)b * C * NP1 + n0 + lanelo;        // col n0
    const float* Xc1 = Xc0 + 16;                                     // col n0+16
    const float* Wr0 = W + (size_t)(ob + 0 * 16 + lanelo) * C;
    const float* Wr1 = W + (size_t)(ob + 1 * 16 + lanelo) * C;
    const float* Wr2 = W + (size_t)(ob + 2 * 16 + lanelo) * C;
    const float* Wr3 = W + (size_t)(ob + 3 * 16 + lanelo) * C;
    float* Yb = Y + (size_t)b * OCH * NP1;

    v8f acc[4][2];
#pragma unroll
    for (int r = 0; r < 4; ++r) {
#pragma unroll
        for (int j = 0; j < 8; ++j) {
            const float bv = bias[ob + r * 16 + j + 8 * half];
            acc[r][0][j] = bv;
            acc[r][1][j] = bv;
        }
    }

#pragma unroll 4
    for (int kk = 0; kk < C; kk += 4) {
        const int kA = kk + 2 * half;
        // ---- issue ALL loads for this K-step (distinct registers) ----
        float xa0 = Xc0[(size_t)kA * NP1];
        float xa1 = Xc0[(size_t)(kA + 1) * NP1];
        float xb0 = Xc1[(size_t)kA * NP1];
        float xb1 = Xc1[(size_t)(kA + 1) * NP1];
        v2f a0, a1, a2, a3;
        a0.x = Wr0[kA]; a0.y = Wr0[kA + 1];
        a1.x = Wr1[kA]; a1.y = Wr1[kA + 1];
        a2.x = Wr2[kA]; a2.y = Wr2[kA + 1];
        a3.x = Wr3[kA]; a3.y = Wr3[kA + 1];
        if (FUSED) {
            const float s0 = s[kA], s1 = s[kA + 1];
            const float t0 = t[kA], t1 = t[kA + 1];
            xa0 = fmaxf(fmaf(s0, xa0, t0), 0.0f);
            xa1 = fmaxf(fmaf(s1, xa1, t1), 0.0f);
            xb0 = fmaxf(fmaf(s0, xb0, t0), 0.0f);
            xb1 = fmaxf(fmaf(s1, xb1, t1), 0.0f);
        }
        v2f bf0; bf0.x = xa0; bf0.y = xa1;
        v2f bf1; bf1.x = xb0; bf1.y = xb1;
        // ---- WMMA burst (8x V_WMMA_F32_16X16X4_F32) ----
        acc[0][0] = __builtin_amdgcn_wmma_f32_16x16x4_f32(
            false, a0, false, bf0, (short)0, acc[0][0], false, false);
        acc[0][1] = __builtin_amdgcn_wmma_f32_16x16x4_f32(
            false, a0, false, bf1, (short)0, acc[0][1], false, false);
        acc[1][0] = __builtin_amdgcn_wmma_f32_16x16x4_f32(
            false, a1, false, bf0, (short)0, acc[1][0], false, false);
        acc[1][1] = __builtin_amdgcn_wmma_f32_16x16x4_f32(
            false, a1, false, bf1, (short)0, acc[1][1], false, false);
        acc[2][0] = __builtin_amdgcn_wmma_f32_16x16x4_f32(
            false, a2, false, bf0, (short)0, acc[2][0], false, false);
        acc[2][1] = __builtin_amdgcn_wmma_f32_16x16x4_f32(
            false, a2, false, bf1, (short)0, acc[2][1], false, false);
        acc[3][0] = __builtin_amdgcn_wmma_f32_16x16x4_f32(
            false, a3, false, bf0, (short)0, acc[3][0], false, false);
        acc[3][1] = __builtin_amdgcn_wmma_f32_16x16x4_f32(
            false, a3, false, bf1, (short)0, acc[3][1], false, false);
    }

#pragma unroll
    for (int r = 0; r < 4; ++r) {
#pragma unroll
        for (int j = 0; j < 8; ++j) {
            const size_t row = (size_t)(ob + r * 16 + j + 8 * half) * NP1;
            Yb[row + n0 + lanelo]      = acc[r][0][j];
            Yb[row + n0 + 16 + lanelo] = acc[r][1][j];
        }
    }
}

// ---------------------------------------------------------------------------
// Kernel 4: per-channel batch stats over (B, N1) -> fold into scale/shift.
// One block (256 threads) per channel.
// ---------------------------------------------------------------------------
__global__ void fp_bnstats_kernel(const float* __restrict__ Y,
                                  const float* __restrict__ gamma,
                                  const float* __restrict__ beta,
                                  float* __restrict__ s,
                                  float* __restrict__ t) {
    __shared__ float ssum[256];
    __shared__ float ssq[256];
    const int o = blockIdx.x;
    float sum = 0.0f, sq = 0.0f;
    for (int b = 0; b < BATCH; ++b) {
        const float* row = Y + ((size_t)b * OCH + o) * NP1;
        for (int n = threadIdx.x; n < NP1; n += 256) {
            const float v = row[n];
            sum += v;
            sq = fmaf(v, v, sq);
        }
    }
    ssum[threadIdx.x] = sum;
    ssq[threadIdx.x] = sq;
    __syncthreads();
    for (int st = 128; st > 0; st >>= 1) {
        if (threadIdx.x < st) {
            ssum[threadIdx.x] += ssum[threadIdx.x + st];
            ssq[threadIdx.x]  += ssq[threadIdx.x + st];
        }
        __syncthreads();
    }
    if (threadIdx.x == 0) {
        const float inv = 1.0f / (float)(BATCH * NP1);
        const float mu = ssum[0] * inv;
        const float var = fmaf(-mu, mu, ssq[0] * inv);
        const float sc = gamma[o] * rsqrtf(var + BN_EPS);
        s[o] = sc;
        t[o] = fmaf(-mu, sc, beta[o]);
    }
}

// ---------------------------------------------------------------------------
// Kernel 5: in-place BN + ReLU on the final output.
// ---------------------------------------------------------------------------
__global__ void fp_bnrelu_kernel(float* __restrict__ Y,
                                 const float* __restrict__ s,
                                 const float* __restrict__ t) {
    const int i = blockIdx.x * 256 + threadIdx.x;
    const int o = (i / NP1) % OCH;
    Y[i] = fmaxf(fmaf(s[o], Y[i], t[o]), 0.0f);
}

// ---------------------------------------------------------------------------
extern "C" void kernel_launch(void* const* d_in, const int* in_sizes, int n_in,
                              void* d_out, int out_size, void* d_ws, size_t ws_size,
                              hipStream_t stream) {
    const float* p1  = (const float*)d_in[0];
    const float* x1  = (const float*)d_in[1];
    const float* p2  = (const float*)d_in[2];
    const float* x2  = (const float*)d_in[3];
    const float* W1  = (const float*)d_in[4];
    const float* b1  = (const float*)d_in[5];
    const float* g1  = (const float*)d_in[6];
    const float* be1 = (const float*)d_in[7];
    const float* W2  = (const float*)d_in[8];
    const float* b2  = (const float*)d_in[9];
    const float* g2  = (const float*)d_in[10];
    const float* be2 = (const float*)d_in[11];
    float* out = (float*)d_out;

    // Workspace layout (bytes): idx | w | X0 | Y1 | s1 t1 s2 t2  (~85 MB)
    char* ws = (char*)d_ws;
    int*   idx = (int*)ws;                                    // B*N1*3 ints
    float* w   = (float*)(ws + (size_t)BATCH * NP1 * 3 * 4);  // B*N1*3 floats
    float* X0  = (float*)(ws + (size_t)BATCH * NP1 * 3 * 8);  // B*384*N1
    float* Y1  = X0 + (size_t)BATCH * C0 * NP1;               // B*256*N1
    float* s1  = Y1 + (size_t)BATCH * OCH * NP1;
    float* t1  = s1 + OCH;
    float* s2  = t1 + OCH;
    float* t2  = s2 + OCH;

    // 1) 3-NN indices + weights
    fp_knn_kernel<<<dim3(NP1 / 256, BATCH), 256, 0, stream>>>(p1, p2, idx, w);

    // 2) X0 = concat([x1, interp(x2)])
    {
        const int total = BATCH * C0 * NP1;
        fp_interp_concat_kernel<<<total / 256, 256, 0, stream>>>(x1, x2, idx, w, X0);
    }

    // 3) GEMM1: Y1 = W1 * X0 + b1   (pre-BN)
    fp_gemm_wmma_kernel<false><<<dim3(NP1 / 256, OCH / 64, BATCH), 256, 0, stream>>>(
        W1, b1, X0, Y1, nullptr, nullptr, C0);

    // 4) BN1 stats -> (s1, t1)
    fp_bnstats_kernel<<<OCH, 256, 0, stream>>>(Y1, g1, be1, s1, t1);

    // 5) GEMM2 with fused BN1+ReLU on the input: out(pre-BN2) = W2*relu(bn(Y1)) + b2
    fp_gemm_wmma_kernel<true><<<dim3(NP1 / 256, OCH / 64, BATCH), 256, 0, stream>>>(
        W2, b2, Y1, out, s1, t1, OCH);

    // 6) BN2 stats -> (s2, t2)
    fp_bnstats_kernel<<<OCH, 256, 0, stream>>>(out, g2, be2, s2, t2);

    // 7) in-place BN2 + ReLU on d_out
    fp_bnrelu_kernel<<<(BATCH * OCH * NP1) / 256, 256, 0, stream>>>(out, s2, t2);
}